// GCN_2946347565080
// MI455X (gfx1250) — compile-verified
//
#include <hip/hip_runtime.h>
#include <hip/hip_bf16.h>
#include <math.h>

// GCN 2-layer inference for MI455X (gfx1250, wave32, WMMA, TDM).
// Pipeline: deg -> dinv -> h1 = x@W1 (WMMA f32 16x16x4, W1 staged via
// tensor_load_to_lds) -> agg1 (atomics) -> relu+bias+W2 -> agg2 into d_out
// -> log_softmax in place.

typedef __attribute__((ext_vector_type(2))) float v2f;
typedef __attribute__((ext_vector_type(8))) float v8f;
typedef __attribute__((ext_vector_type(4))) unsigned int u32x4;
typedef __attribute__((ext_vector_type(4))) int i32x4;
typedef __attribute__((ext_vector_type(8))) int i32x8;

#define D_FEAT    512
#define HIDDEN    16
#define NCLS      7
#define TPAD      8           // padded stride for per-node class vector

#if defined(__has_builtin)
#if __has_builtin(__builtin_amdgcn_tensor_load_to_lds) && \
    __has_builtin(__builtin_amdgcn_s_wait_tensorcnt)
#define USE_TDM 1
#endif
#endif

// ---------------- Kernel A: h1 = x @ W1 via V_WMMA_F32_16X16X4_F32 ----------
// One wave computes a 16x16 output tile (16 nodes x HIDDEN). A is fed from
// global (streaming, 205 MB total). B (W1, 512x16 f32 = 32 KB) is DMA'd
// row-major into LDS once per block by wave 0 via the Tensor Data Mover.
__global__ __launch_bounds__(256) void gemm1_wmma(const float* __restrict__ x,
                                                  const float* __restrict__ W1,
                                                  float* __restrict__ h1,
                                                  int n_nodes) {
    __shared__ float ldsW[D_FEAT * HIDDEN];   // row-major W1[k][n]

#ifdef USE_TDM
    if (threadIdx.x < 32) {                    // wave 0 issues the TDM DMA
        unsigned long long ga = (unsigned long long)(const void*)W1;
        unsigned lds_off = (unsigned)(unsigned long long)(const void*)ldsW;
        u32x4 g0;
        g0.x = 1u;                                         // count=1, user D#
        g0.y = lds_off;                                    // lds_addr [63:32]
        g0.z = (unsigned)(ga & 0xFFFFFFFFull);             // global_addr lo
        g0.w = (unsigned)((ga >> 32) & 0x1FFFFFFull)       // global_addr hi
             | (2u << 30);                                 // type=2 ("image")
        i32x8 g1;
        g1.s0 = 0x00020000;      // data_size=2 -> 4-byte elements
        g1.s1 = (int)(16u  << 16);  // tensor_dim0 = 16   (bits 79:48 lo half)
        g1.s2 = (int)(512u << 16);  // tensor_dim1 = 512  (bits 111:80 lo half)
        g1.s3 = (int)(16u  << 16);  // tile_dim0 = 16     (bits 127:112)
        g1.s4 = 512;                // tile_dim1 = 512    (bits 143:128)
        g1.s5 = 16;                 // tensor_dim0_stride = 16 (bits 207:160)
        g1.s6 = 0;                  // tensor_dim1_stride = 0 (unused, 2-D)
        g1.s7 = 0;
        i32x4 g2 = {0, 0, 0, 0};    // 2-D tensor: groups 2/3 zero
        i32x4 g3 = {0, 0, 0, 0};
#if __clang_major__ >= 23
        i32x8 g4 = {0, 0, 0, 0, 0, 0, 0, 0};
        __builtin_amdgcn_tensor_load_to_lds(g0, g1, g2, g3, g4, 0);
#else
        __builtin_amdgcn_tensor_load_to_lds(g0, g1, g2, g3, 0);
#endif
        __builtin_amdgcn_s_wait_tensorcnt(0);
    }
#else
    for (int i = threadIdx.x; i < D_FEAT * HIDDEN; i += blockDim.x)
        ldsW[i] = W1[i];
#endif
    __syncthreads();

    const int wave = threadIdx.x >> 5;
    const int lane = threadIdx.x & 31;
    const int tile = blockIdx.x * 8 + wave;
    const int row_base = tile * 16;
    if (row_base >= n_nodes) return;

    const int half = lane >> 4;     // 0: lanes 0-15, 1: lanes 16-31
    const int l16  = lane & 15;
    const int koff = half << 1;     // A/B K-offset per ISA layout: 0 or 2
    int arow = row_base + l16;
    if (arow >= n_nodes) arow = n_nodes - 1;   // clamp (harmless, tile guarded)
    const float* __restrict__ xrow = x + (size_t)arow * D_FEAT + koff;
    // row-major W1 in LDS: element (k, n) at ldsW[k*16 + n]
    const float* __restrict__ wptr = ldsW + (size_t)koff * HIDDEN + l16;

    v8f acc = {};
#pragma unroll 8
    for (int k0 = 0; k0 < D_FEAT; k0 += 4) {
        v2f a, b;
        a.x = xrow[k0];              a.y = xrow[k0 + 1];        // b64 global
        b.x = wptr[k0 * HIDDEN];     b.y = wptr[k0 * HIDDEN + HIDDEN];
        acc = __builtin_amdgcn_wmma_f32_16x16x4_f32(
            /*neg_a=*/false, a, /*neg_b=*/false, b,
            /*c_mod=*/(short)0, acc, /*reuse_a=*/false, /*reuse_b=*/false);
    }

    // C/D layout: VGPR r -> M = r + 8*half, N = l16
    if (row_base + 16 <= n_nodes) {
#pragma unroll
        for (int r = 0; r < 8; ++r)
            h1[(size_t)(row_base + r + half * 8) * HIDDEN + l16] = acc[r];
    } else {
#pragma unroll
        for (int r = 0; r < 8; ++r) {
            int m = row_base + r + half * 8;
            if (m < n_nodes) h1[(size_t)m * HIDDEN + l16] = acc[r];
        }
    }
}

// ---------------- degree / normalization ------------------------------------
__global__ void deg_init(float* __restrict__ deg, int n) {
    int i = blockIdx.x * blockDim.x + threadIdx.x;
    if (i < n) deg[i] = 1.0f;                 // self loop
}

__global__ void deg_edges(const long long* __restrict__ dst,
                          float* __restrict__ deg, int n_edges) {
    int e = blockIdx.x * blockDim.x + threadIdx.x;
    if (e < n_edges) atomicAdd(&deg[(int)dst[e]], 1.0f);
}

__global__ void make_dinv(float* __restrict__ deg, int n) {
    int i = blockIdx.x * blockDim.x + threadIdx.x;
    if (i < n) {
        float d = deg[i];
        deg[i] = (d > 0.0f) ? rsqrtf(d) : 0.0f;
    }
}

// ---------------- layer-1 aggregation ---------------------------------------
__global__ void agg1_init(const float* __restrict__ h1,
                          const float* __restrict__ dinv,
                          float* __restrict__ agg1, int total) {
    int i = blockIdx.x * blockDim.x + threadIdx.x;   // over n*HIDDEN
    if (i < total) {
        float di = dinv[i >> 4];
        agg1[i] = di * di * h1[i];                   // self-loop term
    }
}

__global__ void agg1_edges(const long long* __restrict__ src,
                           const long long* __restrict__ dst,
                           const float* __restrict__ h1,
                           const float* __restrict__ dinv,
                           float* __restrict__ agg1, int n_edges) {
    int gid = blockIdx.x * blockDim.x + threadIdx.x; // 16 lanes per edge
    int e = gid >> 4, c = gid & 15;
    if (e < n_edges) {
        int s = (int)src[e], d = (int)dst[e];
        float norm = dinv[s] * dinv[d];
        atomicAdd(&agg1[(size_t)d * HIDDEN + c], h1[(size_t)s * HIDDEN + c] * norm);
    }
}

// ---------------- bias + ReLU + tiny GEMM (16x7) ----------------------------
__global__ __launch_bounds__(256) void relu_gemm2(const float* __restrict__ agg1,
                                                  const float* __restrict__ b1,
                                                  const float* __restrict__ W2,
                                                  float* __restrict__ t, int n) {
    __shared__ float sW2[HIDDEN * NCLS];
    __shared__ float sb1[HIDDEN];
    if (threadIdx.x < HIDDEN * NCLS) sW2[threadIdx.x] = W2[threadIdx.x];
    if (threadIdx.x < HIDDEN)        sb1[threadIdx.x] = b1[threadIdx.x];
    __syncthreads();

    int i = blockIdx.x * blockDim.x + threadIdx.x;
    if (i >= n) return;
    float hreg[HIDDEN];
#pragma unroll
    for (int k = 0; k < HIDDEN; ++k)
        hreg[k] = fmaxf(agg1[(size_t)i * HIDDEN + k] + sb1[k], 0.0f);
#pragma unroll
    for (int c = 0; c < NCLS; ++c) {
        float s = 0.0f;
#pragma unroll
        for (int k = 0; k < HIDDEN; ++k) s = fmaf(hreg[k], sW2[k * NCLS + c], s);
        t[(size_t)i * TPAD + c] = s;
    }
}

// ---------------- layer-2 aggregation (into d_out) --------------------------
__global__ void agg2_init(const float* __restrict__ t,
                          const float* __restrict__ dinv,
                          const float* __restrict__ b2,
                          float* __restrict__ out, int n) {
    int i = blockIdx.x * blockDim.x + threadIdx.x;
    if (i >= n) return;
    float di = dinv[i], n2 = di * di;
#pragma unroll
    for (int c = 0; c < NCLS; ++c)
        out[(size_t)i * NCLS + c] = n2 * t[(size_t)i * TPAD + c] + b2[c];
}

__global__ void agg2_edges(const long long* __restrict__ src,
                           const long long* __restrict__ dst,
                           const float* __restrict__ t,
                           const float* __restrict__ dinv,
                           float* __restrict__ out, int n_edges) {
    int gid = blockIdx.x * blockDim.x + threadIdx.x; // 8 lanes per edge
    int e = gid >> 3, c = gid & 7;
    if (e < n_edges && c < NCLS) {
        int s = (int)src[e], d = (int)dst[e];
        float norm = dinv[s] * dinv[d];
        atomicAdd(&out[(size_t)d * NCLS + c], t[(size_t)s * TPAD + c] * norm);
    }
}

// ---------------- log_softmax over 7 classes, in place ----------------------
__global__ void log_softmax7(float* __restrict__ out, int n) {
    int i = blockIdx.x * blockDim.x + threadIdx.x;
    if (i >= n) return;
    float v[NCLS];
    float m = -INFINITY;
#pragma unroll
    for (int c = 0; c < NCLS; ++c) { v[c] = out[(size_t)i * NCLS + c]; m = fmaxf(m, v[c]); }
    float s = 0.0f;
#pragma unroll
    for (int c = 0; c < NCLS; ++c) s += __expf(v[c] - m);
    float lse = m + __logf(s);
#pragma unroll
    for (int c = 0; c < NCLS; ++c) out[(size_t)i * NCLS + c] = v[c] - lse;
}

// ---------------- launch ----------------------------------------------------
extern "C" void kernel_launch(void* const* d_in, const int* in_sizes, int n_in,
                              void* d_out, int out_size, void* d_ws, size_t ws_size,
                              hipStream_t stream) {
    const float*     x  = (const float*)d_in[0];
    const long long* ei = (const long long*)d_in[1];   // int64 (2, E)
    const float*     W1 = (const float*)d_in[2];
    const float*     b1 = (const float*)d_in[3];
    const float*     W2 = (const float*)d_in[4];
    const float*     b2 = (const float*)d_in[5];
    float* out = (float*)d_out;

    const int n_nodes = in_sizes[0] / D_FEAT;
    const int n_edges = in_sizes[1] / 2;
    const long long* src = ei;
    const long long* dst = ei + n_edges;

    // workspace: h1[n*16] | dinv[n] | agg1[n*16] | t[n*8]  (~16.4 MB)
    float* ws   = (float*)d_ws;
    float* h1   = ws;
    float* dinv = h1   + (size_t)n_nodes * HIDDEN;
    float* agg1 = dinv + (size_t)n_nodes;
    float* tbuf = agg1 + (size_t)n_nodes * HIDDEN;

    const int B = 256;
    auto cdiv = [](long long a, long long b) { return (int)((a + b - 1) / b); };

    deg_init  <<<cdiv(n_nodes, B), B, 0, stream>>>(dinv, n_nodes);
    deg_edges <<<cdiv(n_edges, B), B, 0, stream>>>(dst, dinv, n_edges);
    make_dinv <<<cdiv(n_nodes, B), B, 0, stream>>>(dinv, n_nodes);

    const int tiles = cdiv(n_nodes, 16);
    gemm1_wmma<<<cdiv(tiles, 8), B, 0, stream>>>(x, W1, h1, n_nodes);

    agg1_init <<<cdiv((long long)n_nodes * HIDDEN, B), B, 0, stream>>>(h1, dinv, agg1,
                                                                       n_nodes * HIDDEN);
    agg1_edges<<<cdiv((long long)n_edges * 16, B), B, 0, stream>>>(src, dst, h1, dinv,
                                                                   agg1, n_edges);

    relu_gemm2<<<cdiv(n_nodes, B), B, 0, stream>>>(agg1, b1, W2, tbuf, n_nodes);

    agg2_init <<<cdiv(n_nodes, B), B, 0, stream>>>(tbuf, dinv, b2, out, n_nodes);
    agg2_edges<<<cdiv((long long)n_edges * 8, B), B, 0, stream>>>(src, dst, tbuf, dinv,
                                                                  out, n_edges);

    log_softmax7<<<cdiv(n_nodes, B), B, 0, stream>>>(out, n_nodes);
}